// CalibratedISP_2654289789230
// MI455X (gfx1250) — compile-verified
//
#include <hip/hip_runtime.h>
#include <math.h>

// ---------------------------------------------------------------------------
// CalibratedISP for MI455X (gfx1250)
//
// Roofline: 25.2M pixels * 24B = ~604MB HBM traffic -> ~26us at 23.3 TB/s.
// ~30 VALU ops/pixel -> memory bound. Strategy: double-buffered async
// global->LDS tile streaming (gfx1250 GLOBAL_LOAD_ASYNC_TO_LDS_B128 on
// ASYNCcnt), med3 clamps, fma-form piecewise-linear LUT in LDS, and
// non-temporal 128-bit stores (write-once output, spare the 192MB L2).
// ---------------------------------------------------------------------------

#define TPB        256
#define PPT        4                       // pixels per thread
#define TILE_PIX   (TPB * PPT)             // 1024 pixels / tile
#define TILE_FLTS  (TILE_PIX * 3)          // 3072 floats
#define TILE_BYTES (TILE_FLTS * 4)         // 12288 bytes
#define MAX_GRID   3072

typedef float v4f __attribute__((ext_vector_type(4)));

// --------------------------- device helpers --------------------------------
__device__ __forceinline__ float clamp01(float v) {
  return fminf(fmaxf(v, 0.f), 1.f);        // folds to v_med3_f32
}

// gfx1250 async copy: global (SADDR + 32b VADDR byte offset) -> LDS.
// LDS byte address = low 32 bits of the generic pointer (ISA 10.2: flat LDS
// addresses carry the wave-relative LDS offset in addr[31:0]).
__device__ __forceinline__ void async_tile_b128(const float* gbase,
                                                unsigned goff_bytes,
                                                const float* lds_dst) {
  unsigned lds_addr = (unsigned)(unsigned long long)lds_dst;
  unsigned long long base = (unsigned long long)gbase;
  asm volatile("global_load_async_to_lds_b128 %0, %1, %2"
               :
               : "v"(lds_addr), "v"(goff_bytes), "s"(base)
               : "memory");
}

__device__ __forceinline__ void wait_async_le3() {
  asm volatile("s_wait_asynccnt 0x3" ::: "memory");
}
__device__ __forceinline__ void wait_async_le0() {
  asm volatile("s_wait_asynccnt 0x0" ::: "memory");
}

// ------------------------------ main kernel --------------------------------
// ws layout (floats):
//   [0..8]    M' = diag(T) * M  (row major)
//   [9..11]   b
//   [16..111] 48 x float2 {slope, A} indexed by (k*3 + c)
//             out(y) = clamp01( fma(slope, y, A) ),  A = cum[k] - slope*k/16
__global__ void __launch_bounds__(TPB)
isp_main_kernel(const float* __restrict__ x, const float* __restrict__ wsf,
                float* __restrict__ out, int pixels) {
  __shared__ __align__(16) float tile[2][TILE_FLTS];   // 2 x 12KB
  __shared__ float2 lut[48];

  const int tid = threadIdx.x;
  if (tid < 48) lut[tid] = ((const float2*)(wsf + 16))[tid];

  // Uniform constants -> scalar loads
  const float m00 = wsf[0], m01 = wsf[1], m02 = wsf[2];
  const float m10 = wsf[3], m11 = wsf[4], m12 = wsf[5];
  const float m20 = wsf[6], m21 = wsf[7], m22 = wsf[8];
  const float bb0 = wsf[9], bb1 = wsf[10], bb2 = wsf[11];

  const int fullTiles = pixels / TILE_PIX;
  const int stride = (int)gridDim.x;

  if ((int)blockIdx.x < fullTiles) {
    // Prologue: issue first tile into buffer 0 (3 x b128 per thread).
    {
      const unsigned g0 = (unsigned)blockIdx.x * TILE_BYTES + (unsigned)tid * 16u;
      async_tile_b128(x, g0,          &tile[0][tid * 4]);
      async_tile_b128(x, g0 + 4096u,  &tile[0][tid * 4 + 1024]);
      async_tile_b128(x, g0 + 8192u,  &tile[0][tid * 4 + 2048]);
    }
    int buf = 0;
    for (int t = (int)blockIdx.x; t < fullTiles; t += stride) {
      const int tn = t + stride;
      if (tn < fullTiles) {
        const int nb = buf ^ 1;
        const unsigned g0 = (unsigned)tn * TILE_BYTES + (unsigned)tid * 16u;
        async_tile_b128(x, g0,         &tile[nb][tid * 4]);
        async_tile_b128(x, g0 + 4096u, &tile[nb][tid * 4 + 1024]);
        async_tile_b128(x, g0 + 8192u, &tile[nb][tid * 4 + 2048]);
        if (tn + stride < fullTiles)   // L2 prefetch two tiles ahead
          __builtin_prefetch(x + (size_t)(tn + stride) * TILE_FLTS + tid * 4, 0, 1);
        wait_async_le3();              // current tile's 3 loads are done
      } else {
        wait_async_le0();
      }
      __syncthreads();                 // tile (and LUT) visible to all waves

      const float* src = &tile[buf][tid * 12];
      const v4f va = ((const v4f*)src)[0];
      const v4f vb = ((const v4f*)src)[1];
      const v4f vc = ((const v4f*)src)[2];
      const float in[12] = {va.x, va.y, va.z, va.w,
                            vb.x, vb.y, vb.z, vb.w,
                            vc.x, vc.y, vc.z, vc.w};
      float o[12];
#pragma unroll
      for (int p = 0; p < PPT; ++p) {
        const float r = in[3 * p], gch = in[3 * p + 1], bl = in[3 * p + 2];
        const float y0 = clamp01(fmaf(m00, r, fmaf(m01, gch, fmaf(m02, bl, bb0))));
        const float y1 = clamp01(fmaf(m10, r, fmaf(m11, gch, fmaf(m12, bl, bb1))));
        const float y2 = clamp01(fmaf(m20, r, fmaf(m21, gch, fmaf(m22, bl, bb2))));
        int k0 = (int)(y0 * 16.f); k0 = k0 > 15 ? 15 : k0;
        int k1 = (int)(y1 * 16.f); k1 = k1 > 15 ? 15 : k1;
        int k2 = (int)(y2 * 16.f); k2 = k2 > 15 ? 15 : k2;
        const float2 s0 = lut[k0 * 3 + 0];
        const float2 s1 = lut[k1 * 3 + 1];
        const float2 s2 = lut[k2 * 3 + 2];
        o[3 * p]     = clamp01(fmaf(s0.x, y0, s0.y));
        o[3 * p + 1] = clamp01(fmaf(s1.x, y1, s1.y));
        o[3 * p + 2] = clamp01(fmaf(s2.x, y2, s2.y));
      }
      // Write-once output: non-temporal 128-bit stores (th:TH_STORE_NT).
      v4f* gdst = (v4f*)(out + (size_t)t * TILE_FLTS + tid * 12);
      __builtin_nontemporal_store((v4f){o[0], o[1], o[2],  o[3]},  gdst + 0);
      __builtin_nontemporal_store((v4f){o[4], o[5], o[6],  o[7]},  gdst + 1);
      __builtin_nontemporal_store((v4f){o[8], o[9], o[10], o[11]}, gdst + 2);
      __syncthreads();                 // WAR: buffer reusable for next issue
      buf ^= 1;
    }
  }

  __syncthreads();                     // LUT visibility for tail-only blocks

  // Tail (not taken for the reference shape: 25,165,824 % 1024 == 0)
  for (int p = fullTiles * TILE_PIX + (int)blockIdx.x * TPB + tid; p < pixels;
       p += stride * TPB) {
    const float r = x[3 * p], gch = x[3 * p + 1], bl = x[3 * p + 2];
    const float y0 = clamp01(fmaf(m00, r, fmaf(m01, gch, fmaf(m02, bl, bb0))));
    const float y1 = clamp01(fmaf(m10, r, fmaf(m11, gch, fmaf(m12, bl, bb1))));
    const float y2 = clamp01(fmaf(m20, r, fmaf(m21, gch, fmaf(m22, bl, bb2))));
    int k0 = (int)(y0 * 16.f); k0 = k0 > 15 ? 15 : k0;
    int k1 = (int)(y1 * 16.f); k1 = k1 > 15 ? 15 : k1;
    int k2 = (int)(y2 * 16.f); k2 = k2 > 15 ? 15 : k2;
    const float2 s0 = lut[k0 * 3 + 0];
    const float2 s1 = lut[k1 * 3 + 1];
    const float2 s2 = lut[k2 * 3 + 2];
    out[3 * p]     = clamp01(fmaf(s0.x, y0, s0.y));
    out[3 * p + 1] = clamp01(fmaf(s1.x, y1, s1.y));
    out[3 * p + 2] = clamp01(fmaf(s2.x, y2, s2.y));
  }
}

// ---------------- setup: fold T into M, build fma-form LUT -----------------
__global__ void __launch_bounds__(64)
isp_setup_kernel(const float* __restrict__ M, const float* __restrict__ T,
                 const float* __restrict__ b, const float* __restrict__ raw,
                 float* __restrict__ ws) {
  const int t = threadIdx.x;
  if (t < 9) {
    ws[t] = T[t / 3] * M[t];
  } else if (t < 12) {
    ws[t] = b[t - 9];
  } else if (t < 15) {
    const int c = t - 12;
    float mx = -1e30f;
    for (int k = 0; k < 16; ++k) mx = fmaxf(mx, raw[k * 3 + c]);
    float e[16];
    float s = 0.f;
    for (int k = 0; k < 16; ++k) { e[k] = expf(raw[k * 3 + c] - mx); s += e[k]; }
    const float inv = 16.f / s;            // softmax * K
    float cum = 0.f;
    for (int k = 0; k < 16; ++k) {
      const float slope = e[k] * inv;
      ws[16 + (k * 3 + c) * 2 + 0] = slope;
      ws[16 + (k * 3 + c) * 2 + 1] = cum - slope * ((float)k * 0.0625f);
      cum += slope * 0.0625f;              // heights = slope / K
    }
  }
}

// ------------------------------ launcher -----------------------------------
extern "C" void kernel_launch(void* const* d_in, const int* in_sizes, int n_in,
                              void* d_out, int out_size, void* d_ws, size_t ws_size,
                              hipStream_t stream) {
  (void)n_in; (void)out_size; (void)ws_size;
  const float* x   = (const float*)d_in[0];
  const float* M   = (const float*)d_in[1];
  const float* T   = (const float*)d_in[2];
  const float* b   = (const float*)d_in[3];
  const float* raw = (const float*)d_in[4];
  float* out = (float*)d_out;
  float* ws  = (float*)d_ws;

  const int nFloats = in_sizes[0];
  const int pixels  = nFloats / 3;

  isp_setup_kernel<<<1, 64, 0, stream>>>(M, T, b, raw, ws);

  const int fullTiles = pixels / TILE_PIX;
  int grid = fullTiles < MAX_GRID ? fullTiles : MAX_GRID;
  if (grid < 1) grid = 1;
  isp_main_kernel<<<grid, TPB, 0, stream>>>(x, ws, out, pixels);
}